// Model_83141976916164
// MI455X (gfx1250) — compile-verified
//
#include <hip/hip_runtime.h>

// ---------------------------------------------------------------------------
// 6-layer GRU on MI455X (gfx1250), fp32, via V_WMMA_F32_16X16X4_F32.
//
// Design (latency-bound recurrence, see analysis):
//  * One persistent kernel per layer, 32 WGs x 256 threads (8 waves).
//  * Each wave owns a 16(b) x 16(j) tile of h and computes all three gate
//    columns (r,z,n) for it -> gate math is wave-local.
//  * W_hh / W_ih fragments pre-swizzled into LDS in WMMA per-lane layout
//    (ISA 7.12.2): 48KB + 48KB per WG (CDNA5 allows 320KB/WG).
//  * Per-step cross-WG sync: monotonic atomic counter in L2
//    (release fetch_add / acquire spin + s_sleep). h is double-buffered.
//  * gi_{t+1} = x_{t+1} @ Wih^T + bih is computed right after signaling
//    step t, hiding the input GEMM in the sync window.
//
// Workspace requirement: ~270 MB (2 x (T*B*H) fp32 sequence buffers
// + 2 x (B*H) h buffers + 6 counters).
// ---------------------------------------------------------------------------

#define T_SEQ 512
#define BATCH 256
#define INSZ  128
#define HID   256
#define NL    6
#define G3    (3 * HID)        // 768
#define BH    (BATCH * HID)    // 65536
#define NWG   32u

typedef __attribute__((ext_vector_type(2))) float v2f;
typedef __attribute__((ext_vector_type(8))) float v8f;

__device__ __forceinline__ v8f wmma4(v2f a, v2f b, v8f c) {
  // D = A(16x4 f32) * B(4x16 f32) + C(16x16 f32), exact fp32.
  return __builtin_amdgcn_wmma_f32_16x16x4_f32(
      /*neg_a=*/false, a, /*neg_b=*/false, b,
      /*c_mod=*/(short)0, c, /*reuse_a=*/false, /*reuse_b=*/false);
}

__device__ __forceinline__ v8f splat8(float x) {
  v8f v;
#pragma unroll
  for (int i = 0; i < 8; ++i) v[i] = x;
  return v;
}

__device__ __forceinline__ float sigm_f(float x) {
  return 1.0f / (1.0f + __expf(-x));
}
__device__ __forceinline__ float tanh_f(float x) {
  // 1 - 2/(e^{2x}+1): correct limits at +/-inf, no inf/inf.
  float e = __expf(2.0f * x);
  return 1.0f - 2.0f / (e + 1.0f);
}

// Per-lane WMMA fragment index helpers (ISA 7.12.2, 32-bit layouts):
//  A 16x4 : lane L holds row M = b0+(L&15), K = k + 2*(L>>4) + {0,1}
//  B 4x16 : lane L holds col N = (L&15),    K = k + 2*(L>>4) + {0,1}
//  C 16x16: VGPR j of lane L = element (M = j + 8*(L>>4), N = L&15)

__global__ __launch_bounds__(256, 1)
void gru_layer_scan(const float* __restrict__ xin, int kin,      // (T,B,kin)
                    const float* __restrict__ h0l,               // (B,H)
                    const float* __restrict__ wih,               // (3H,kin)
                    const float* __restrict__ whh,               // (3H,H)
                    const float* __restrict__ bih,               // (3H)
                    const float* __restrict__ bhh,               // (3H)
                    float* __restrict__ seq_out,                 // (T,B,H)
                    float* __restrict__ hbuf0,
                    float* __restrict__ hbuf1,
                    float* __restrict__ hfinal,                  // (B,H)
                    unsigned* __restrict__ cnt) {
  __shared__ v2f fragWhh[(HID / 4) * 3 * 32];   // 48 KB
  __shared__ v2f fragWih[(HID / 4) * 3 * 32];   // 48 KB (kin<=256)

  const int tid    = threadIdx.x;
  const int lane   = tid & 31;
  const int wave   = tid >> 5;
  const int wg     = blockIdx.x;                // 0..31
  const int j0     = (wg >> 1) * 16;            // hidden-column tile
  const int b0     = (wg & 1) * 128 + wave * 16;// batch-row tile
  const int nlocal = lane & 15;
  const int khalf  = (lane >> 4) << 1;
  const int kx     = kin / 4;                   // k-iters for input GEMM
  const int kh     = HID / 4;                   // k-iters for recurrent GEMM

  // ---- Stage pre-swizzled B fragments (weights) into LDS, once. ----
  for (int idx = tid; idx < kh * 3 * 32; idx += 256) {
    int l = idx & 31, gate = (idx >> 5) % 3, kit = idx / 96;
    int ks = kit * 4 + ((l >> 4) << 1);
    int g  = gate * HID + j0 + (l & 15);
    v2f w; w.x = whh[g * HID + ks]; w.y = whh[g * HID + ks + 1];
    fragWhh[idx] = w;
  }
  for (int idx = tid; idx < kx * 3 * 32; idx += 256) {
    int l = idx & 31, gate = (idx >> 5) % 3, kit = idx / 96;
    int ks = kit * 4 + ((l >> 4) << 1);
    int g  = gate * HID + j0 + (l & 15);
    v2f w; w.x = wih[g * kin + ks]; w.y = wih[g * kin + ks + 1];
    fragWih[idx] = w;
  }
  __syncthreads();

  const float bihr = bih[0 * HID + j0 + nlocal];
  const float bihz = bih[1 * HID + j0 + nlocal];
  const float bihn = bih[2 * HID + j0 + nlocal];
  const float bhhr = bhh[0 * HID + j0 + nlocal];
  const float bhhz = bhh[1 * HID + j0 + nlocal];
  const float bhhn = bhh[2 * HID + j0 + nlocal];

  // gi_t = x_t @ Wih^T + bih for this wave's tile (3 gate accumulators).
  auto compute_gi = [&](int t, v8f& cr, v8f& cz, v8f& cn) {
    cr = splat8(bihr); cz = splat8(bihz); cn = splat8(bihn);
    const float* arow = xin + (size_t)t * BATCH * kin + (size_t)(b0 + nlocal) * kin;
    for (int kit = 0; kit < kx; ++kit) {
      v2f a = *(const v2f*)(arow + kit * 4 + khalf);
      cr = wmma4(a, fragWih[(kit * 3 + 0) * 32 + lane], cr);
      cz = wmma4(a, fragWih[(kit * 3 + 1) * 32 + lane], cz);
      cn = wmma4(a, fragWih[(kit * 3 + 2) * 32 + lane], cn);
    }
  };

  v8f gir, giz, gin;
  compute_gi(0, gir, giz, gin);

  for (int t = 0; t < T_SEQ; ++t) {
    // Wait until all WGs published h_t (signals of step t-1).
    if (t > 0) {
      if (tid == 0) {
        const unsigned target = NWG * (unsigned)t;
        while (__hip_atomic_load(cnt, __ATOMIC_ACQUIRE,
                                 __HIP_MEMORY_SCOPE_AGENT) < target)
          __builtin_amdgcn_s_sleep(1);
      }
      __syncthreads();
    }
    const float* hsrc = (t == 0) ? h0l : ((t & 1) ? hbuf1 : hbuf0);
    float*       hdst = (t & 1) ? hbuf0 : hbuf1;   // W(t) = hbuf[(t+1)&1]

    // gh = h_t @ Whh^T + bhh  (3 gate accumulators, K = 256)
    v8f ghr = splat8(bhhr), ghz = splat8(bhhz), ghn = splat8(bhhn);
    const float* arow = hsrc + (size_t)(b0 + nlocal) * HID;
    for (int kit = 0; kit < kh; ++kit) {
      v2f a = *(const v2f*)(arow + kit * 4 + khalf);
      ghr = wmma4(a, fragWhh[(kit * 3 + 0) * 32 + lane], ghr);
      ghz = wmma4(a, fragWhh[(kit * 3 + 1) * 32 + lane], ghz);
      ghn = wmma4(a, fragWhh[(kit * 3 + 2) * 32 + lane], ghn);
    }

    // Gate math + h update, all wave-local (C-fragment layout).
    const int j = j0 + nlocal;
#pragma unroll
    for (int jj = 0; jj < 8; ++jj) {
      const int m = b0 + jj + ((lane >> 4) << 3);
      const float hp = hsrc[m * HID + j];
      const float r  = sigm_f(gir[jj] + ghr[jj]);
      const float z  = sigm_f(giz[jj] + ghz[jj]);
      const float nn = tanh_f(gin[jj] + r * ghn[jj]);
      const float hn = (1.0f - z) * nn + z * hp;
      hdst[m * HID + j] = hn;
      seq_out[(size_t)t * BH + m * HID + j] = hn;
      if (t == T_SEQ - 1) hfinal[m * HID + j] = hn;
    }

    // Publish h_{t+1}: all waves done -> release signal from one lane.
    __syncthreads();
    if (tid == 0) {
      __threadfence();
      __hip_atomic_fetch_add(cnt, 1u, __ATOMIC_RELEASE,
                             __HIP_MEMORY_SCOPE_AGENT);
    }
    // Hide next step's input projection in the sync window.
    if (t + 1 < T_SEQ) compute_gi(t + 1, gir, giz, gin);
  }
}

extern "C" void kernel_launch(void* const* d_in, const int* in_sizes, int n_in,
                              void* d_out, int out_size, void* d_ws, size_t ws_size,
                              hipStream_t stream) {
  (void)in_sizes; (void)n_in; (void)out_size; (void)ws_size;
  const float* x     = (const float*)d_in[0];  // (T,B,I)
  const float* h0    = (const float*)d_in[1];  // (NL,B,H)
  const float* w_ih0 = (const float*)d_in[2];  // (3H,I)
  const float* w_ih  = (const float*)d_in[3];  // (NL-1,3H,H)
  const float* w_hh  = (const float*)d_in[4];  // (NL,3H,H)
  const float* b_ih  = (const float*)d_in[5];  // (NL,3H)
  const float* b_hh  = (const float*)d_in[6];  // (NL,3H)
  float* out = (float*)d_out;                  // (NL,B,H)

  char* ws = (char*)d_ws;
  unsigned* cnt  = (unsigned*)ws;              // 6 step counters
  float* hbuf0   = (float*)(ws + 256);
  float* hbuf1   = hbuf0 + BH;
  float* seqA    = (float*)(ws + (1 << 20));
  float* seqB    = seqA + (size_t)T_SEQ * BH;

  hipMemsetAsync(cnt, 0, NL * sizeof(unsigned), stream);  // graph-capturable

  const float* cur_in = x;
  int kin = INSZ;
  float* bufs[2] = {seqA, seqB};
  for (int L = 0; L < NL; ++L) {
    float* so = bufs[L & 1];
    const float* wih = (L == 0) ? w_ih0 : (w_ih + (size_t)(L - 1) * G3 * HID);
    gru_layer_scan<<<dim3(NWG), dim3(256), 0, stream>>>(
        cur_in, kin,
        h0 + (size_t)L * BH,
        wih,
        w_hh + (size_t)L * G3 * HID,
        b_ih + (size_t)L * G3,
        b_hh + (size_t)L * G3,
        so, hbuf0, hbuf1,
        out + (size_t)L * BH,
        cnt + L);
    cur_in = so;
    kin = HID;
  }
}